// GatherHardRegion_89026082111872
// MI455X (gfx1250) — compile-verified
//
#include <hip/hip_runtime.h>
#include <hip/hip_bf16.h>
#include <math.h>

// Problem constants (from reference): feats (8,512,128,128) f32, probs (8,19,128,128) f32
#define B   8
#define K   19
#define C   512
#define N   16384      // 128*128
#define KT  4096       // 0.25 * N
#define SORT_THREADS 1024

// ---------------------------------------------------------------------------
// Kernel 1: per-pixel softmax top-2 margin -> packed 64-bit sort key.
// key = (monotone_u32(margin) << 32) | (0xFFFFFFFF - pixel_index)
// Descending sort on key == descending margin, ascending index on ties
// (matches stable jnp.argsort(-margin)).
// ---------------------------------------------------------------------------
__global__ void margin_key_kernel(const float* __restrict__ probs,
                                  unsigned long long* __restrict__ keys) {
    const unsigned gid = blockIdx.x * blockDim.x + threadIdx.x;   // 0 .. B*N-1
    const unsigned b = gid >> 14;          // / N
    const unsigned n = gid & (N - 1);
    const float* p = probs + (size_t)b * K * N + n;

    float x[K];
#pragma unroll
    for (int c = 0; c < K; ++c) x[c] = p[(size_t)c * N];

    float m1 = -INFINITY, m2 = -INFINITY;
#pragma unroll
    for (int c = 0; c < K; ++c) {
        float hi = fmaxf(x[c], m1);
        float lo = fminf(x[c], m1);
        m1 = hi;
        m2 = fmaxf(m2, lo);
    }
    float s = 0.0f;
#pragma unroll
    for (int c = 0; c < K; ++c) s += expf(x[c] - m1);

    const float margin = (1.0f - expf(m2 - m1)) / s;   // p_top1 - p_top2

    unsigned u = __float_as_uint(margin);
    u = (u & 0x80000000u) ? ~u : (u | 0x80000000u);    // monotone float->uint map
    keys[gid] = ((unsigned long long)u << 32) |
                (unsigned long long)(0xFFFFFFFFu - n);
}

// ---------------------------------------------------------------------------
// Kernel 2: one workgroup per batch; bitonic sort of 16384 u64 keys entirely
// in LDS (128 KB dynamic LDS -- CDNA5's 320 KB/WGP makes this possible).
// Emits descending order; writes top-4096 indices (int to ws for the gather,
// float to d_out hard_region slot).
// ---------------------------------------------------------------------------
__global__ void bitonic_sort_kernel(const unsigned long long* __restrict__ keys,
                                    int* __restrict__ idxbuf,
                                    float* __restrict__ out_region) {
    extern __shared__ unsigned long long sm[];       // 16384 * 8B = 128 KB
    const int b   = blockIdx.x;
    const int tid = threadIdx.x;
    const unsigned long long* kb = keys + (size_t)b * N;

    for (int i = tid; i < N; i += SORT_THREADS) sm[i] = kb[i];
    __syncthreads();

    for (int k = 2; k <= N; k <<= 1) {
        for (int j = k >> 1; j > 0; j >>= 1) {
            for (int i = tid; i < N; i += SORT_THREADS) {
                const int ixj = i ^ j;
                if (ixj > i) {
                    unsigned long long a = sm[i];
                    unsigned long long c = sm[ixj];
                    const bool up = ((i & k) == 0);  // "up" segments sort descending
                    if (up ? (a < c) : (a > c)) { sm[i] = c; sm[ixj] = a; }
                }
            }
            __syncthreads();
        }
    }

    for (int i = tid; i < KT; i += SORT_THREADS) {
        const unsigned idx = 0xFFFFFFFFu - (unsigned)(sm[i] & 0xFFFFFFFFull);
        idxbuf[b * KT + i]     = (int)idx;
        out_region[b * KT + i] = (float)idx;         // reference dtype flattened to f32
    }
}

// ---------------------------------------------------------------------------
// Kernel 3: hard_feat[b,c,j] = feats[b,c,idx[b,j]]  (coalesced writes,
// gathered reads; 64 MB in / 64 MB out).
// ---------------------------------------------------------------------------
__global__ void gather_kernel(const float* __restrict__ feats,
                              const int* __restrict__ idxbuf,
                              float* __restrict__ out_hard) {
    const size_t gid = (size_t)blockIdx.x * blockDim.x + threadIdx.x; // 0..2^24-1
    const int j = (int)(gid & (KT - 1));          // 2^12
    const int c = (int)((gid >> 12) & (C - 1));   // 2^9
    const int b = (int)(gid >> 21);
    const int n = idxbuf[b * KT + j];
    out_hard[gid] = feats[((size_t)b * C + c) * N + n];
}

// ---------------------------------------------------------------------------
// Kernel 4: 256 MB feats pass-through copy using the CDNA5 async global<->LDS
// DMA path (ASYNCcnt-tracked, ISA 15.18.3 ops 98/102). Each lane stages one
// 16B element per slot in LDS: 4 x b128 async loads -> s_wait_asynccnt 0 ->
// 4 x b128 async stores. No VGPR data movement; S_ENDPGM's implicit wait-idle
// covers the trailing stores before LDS dealloc.
// Per block: 256 threads * 4 slots * 16 B = 16 KB tile.
// ---------------------------------------------------------------------------
__global__ void copy_async_kernel(const float4* __restrict__ src,
                                  float4* __restrict__ dst) {
    __shared__ float4 cbuf[SORT_THREADS];            // 16 KB staging
    const unsigned tid  = threadIdx.x;
    const size_t   base = (size_t)blockIdx.x * 1024u;

#pragma unroll
    for (int i = 0; i < 4; ++i) {
        const unsigned slot = i * 256u + tid;
        const unsigned long long gaddr =
            (unsigned long long)(uintptr_t)(src + base + slot);
        // Low 32 bits of the generic shared pointer == LDS byte offset.
        const unsigned lofs = (unsigned)(uintptr_t)&cbuf[slot];
        asm volatile("global_load_async_to_lds_b128 %0, %1, off"
                     :: "v"(lofs), "v"(gaddr) : "memory");
    }

    asm volatile("s_wait_asynccnt 0" ::: "memory");

#pragma unroll
    for (int i = 0; i < 4; ++i) {
        const unsigned slot = i * 256u + tid;
        const unsigned long long gaddr =
            (unsigned long long)(uintptr_t)(dst + base + slot);
        const unsigned lofs = (unsigned)(uintptr_t)&cbuf[slot];
        asm volatile("global_store_async_from_lds_b128 %0, %1, off"
                     :: "v"(gaddr), "v"(lofs) : "memory");
    }
}

// ---------------------------------------------------------------------------
// Launch: d_out = [hard_feat (B*C*KT) | feats copy (B*C*N) | hard_region (B*KT)]
// ws: [0, 1MB) packed u64 keys ; [1MB, 1MB+128KB) int indices.
// ---------------------------------------------------------------------------
extern "C" void kernel_launch(void* const* d_in, const int* in_sizes, int n_in,
                              void* d_out, int out_size, void* d_ws, size_t ws_size,
                              hipStream_t stream) {
    const float* feats = (const float*)d_in[0];
    const float* probs = (const float*)d_in[1];

    float* out        = (float*)d_out;
    float* out_hard   = out;                                  // B*C*KT
    float* out_feats  = out_hard + (size_t)B * C * KT;        // B*C*N
    float* out_region = out_feats + (size_t)B * C * N;        // B*KT

    unsigned long long* keys = (unsigned long long*)d_ws;     // B*N u64 = 1 MB
    int* idxbuf = (int*)((char*)d_ws + (size_t)B * N * sizeof(unsigned long long));

    // 1) margins -> sort keys
    margin_key_kernel<<<(B * N) / 256, 256, 0, stream>>>(probs, keys);

    // 2) per-batch in-LDS bitonic sort (128 KB dynamic LDS per workgroup)
    bitonic_sort_kernel<<<B, SORT_THREADS, N * sizeof(unsigned long long), stream>>>(
        keys, idxbuf, out_region);

    // 3) gather hard features
    gather_kernel<<<(B * C * KT) / 256, 256, 0, stream>>>(feats, idxbuf, out_hard);

    // 4) async-DMA pass-through copy of feats (B*C*N floats = 2^24 float4s)
    copy_async_kernel<<<(B * C * N) / (4 * 256 * 4), 256, 0, stream>>>(
        (const float4*)feats, (float4*)out_feats);
}